// CropConvolution_31911607009796
// MI455X (gfx1250) — compile-verified
//
#include <hip/hip_runtime.h>

typedef float v2f __attribute__((ext_vector_type(2)));
typedef float v8f __attribute__((ext_vector_type(8)));

#define BATCH   64
#define N_FULL  512
#define F_IN    128
#define F_OUT   128
#define CROP0   128
#define CROP1   384
#define CROP_N  256
#define BN_EPS  1e-5f

// ---------------------------------------------------------------------------
// Kernel 1: support[b, m', f] = sum_k input[b, CROP0+m', k] * weight[k, f]
// One workgroup = 8 waves = one 16x128 output tile; A tile staged in LDS.
// ---------------------------------------------------------------------------
__global__ __launch_bounds__(256)
void support_gemm(const float* __restrict__ input,
                  const float* __restrict__ weight,
                  float* __restrict__ sup) {
    __shared__ float ldsA[16 * 128];                 // 8 KB
    const int bidx = blockIdx.x >> 4;                // batch
    const int m0   = (blockIdx.x & 15) << 4;         // row tile within crop
    const int tid  = threadIdx.x;

    const float* ag = input + ((size_t)bidx * N_FULL + CROP0 + m0) * F_IN;
    for (int i = tid; i < 16 * 128; i += 256) {      // coalesced stage
        int r = i >> 7, c = i & 127;
        ldsA[i] = ag[(size_t)r * F_IN + c];
    }
    __syncthreads();

    const int wave = tid >> 5;
    const int lane = tid & 31;
    const int n0   = wave << 4;                      // 8 waves cover N=0..127
    const int lm   = lane & 15;
    const int half = lane >> 4;                      // K-half select per ISA layout

    v8f acc = {};
    for (int k = 0; k < 128; k += 4) {
        // A 16x4 f32: lanes 0-15 hold K=k,k+1 ; lanes 16-31 hold K=k+2,k+3
        v2f a = *(const v2f*)&ldsA[lm * 128 + k + 2 * half];
        // B 4x16 f32: mirrored layout over N
        v2f b;
        b.x = weight[(size_t)(k + 2 * half)     * F_OUT + n0 + lm];
        b.y = weight[(size_t)(k + 2 * half + 1) * F_OUT + n0 + lm];
        acc = __builtin_amdgcn_wmma_f32_16x16x4_f32(false, a, false, b,
                                                    (short)0, acc, false, false);
    }

    // C/D layout: VGPR r -> M = r + 8*half, N = lane&15
    float* og = sup + ((size_t)bidx * CROP_N + m0) * F_OUT + n0 + lm;
    #pragma unroll
    for (int r = 0; r < 8; ++r) {
        int M = r + 8 * half;
        og[(size_t)M * F_OUT] = acc[r];
    }
}

// ---------------------------------------------------------------------------
// Kernel 2: raw_out[b, CROP0+m', f] = sum_{k<256} adj[b,CROP0+m',CROP0+k] *
//                                     sup[b, k, f]
// Writes raw (pre-BN) values into the crop region of d_out.
// ---------------------------------------------------------------------------
__global__ __launch_bounds__(256)
void aggregate_gemm(const float* __restrict__ adj,
                    const float* __restrict__ sup,
                    float* __restrict__ out) {
    __shared__ float ldsA[16 * 256];                 // 16 KB
    const int bidx = blockIdx.x >> 4;
    const int m0   = (blockIdx.x & 15) << 4;
    const int tid  = threadIdx.x;

    const float* ag = adj + ((size_t)bidx * N_FULL + CROP0 + m0) * N_FULL + CROP0;
    for (int i = tid; i < 16 * 256; i += 256) {
        int r = i >> 8, c = i & 255;
        ldsA[i] = ag[(size_t)r * N_FULL + c];
    }
    __syncthreads();

    const int wave = tid >> 5;
    const int lane = tid & 31;
    const int n0   = wave << 4;
    const int lm   = lane & 15;
    const int half = lane >> 4;

    const float* bg = sup + (size_t)bidx * CROP_N * F_OUT + n0 + lm;
    v8f acc = {};
    for (int k = 0; k < 256; k += 4) {
        v2f a = *(const v2f*)&ldsA[lm * 256 + k + 2 * half];
        v2f b;
        b.x = bg[(size_t)(k + 2 * half)     * F_OUT];
        b.y = bg[(size_t)(k + 2 * half + 1) * F_OUT];
        acc = __builtin_amdgcn_wmma_f32_16x16x4_f32(false, a, false, b,
                                                    (short)0, acc, false, false);
    }

    float* og = out + ((size_t)bidx * N_FULL + CROP0 + m0) * F_OUT + n0 + lm;
    #pragma unroll
    for (int r = 0; r < 8; ++r) {
        int M = r + 8 * half;
        og[(size_t)M * F_OUT] = acc[r];
    }
}

// ---------------------------------------------------------------------------
// Kernel 3: batch-norm stats per crop feature (n, f): mean and rsqrt(var+eps)
// One thread per feature, coalesced over f; 64-sample reduction over batch.
// ---------------------------------------------------------------------------
__global__ __launch_bounds__(256)
void bn_stats(const float* __restrict__ raw,
              float* __restrict__ mean,
              float* __restrict__ rstd) {
    const int idx = blockIdx.x * 256 + threadIdx.x;  // 0 .. 32767
    const int n   = CROP0 + (idx >> 7);
    const int f   = idx & 127;
    const float* p = raw + (size_t)n * F_OUT + f;
    float s = 0.0f, s2 = 0.0f;
    #pragma unroll 4
    for (int b = 0; b < BATCH; ++b) {
        float x = p[(size_t)b * N_FULL * F_OUT];
        s  += x;
        s2 += x * x;
    }
    const float m = s * (1.0f / BATCH);
    float v = s2 * (1.0f / BATCH) - m * m;
    v = fmaxf(v, 0.0f);
    mean[idx] = m;
    rstd[idx] = rsqrtf(v + BN_EPS);
}

// ---------------------------------------------------------------------------
// Kernel 4: in-place normalize; non-crop rows are exactly beta (x=0,mean=0).
// ---------------------------------------------------------------------------
__global__ __launch_bounds__(256)
void bn_apply(float* __restrict__ out,
              const float* __restrict__ mean,
              const float* __restrict__ rstd,
              const float* __restrict__ gamma,
              const float* __restrict__ beta) {
    const int idx = blockIdx.x * 256 + threadIdx.x;  // < 64*512*128
    const int rem = idx & (N_FULL * F_OUT - 1);      // n*128 + f (feature index)
    const int n   = rem >> 7;
    if (n >= CROP0 && n < CROP1) {
        const int cidx = ((n - CROP0) << 7) | (rem & 127);
        const float x = out[idx];
        out[idx] = (x - mean[cidx]) * rstd[cidx] * gamma[rem] + beta[rem];
    } else {
        out[idx] = beta[rem];
    }
}

// ---------------------------------------------------------------------------
extern "C" void kernel_launch(void* const* d_in, const int* in_sizes, int n_in,
                              void* d_out, int out_size, void* d_ws, size_t ws_size,
                              hipStream_t stream) {
    (void)in_sizes; (void)n_in; (void)out_size; (void)ws_size;

    const float* input  = (const float*)d_in[0];   // [64, 512, 128]
    const float* adj    = (const float*)d_in[1];   // [64, 512, 512]
    const float* weight = (const float*)d_in[2];   // [128, 128]
    const float* gamma  = (const float*)d_in[3];   // [65536]
    const float* beta   = (const float*)d_in[4];   // [65536]
    float* out = (float*)d_out;                    // [64, 512, 128]

    // workspace layout: support (8 MB) | mean (128 KB) | rstd (128 KB)
    float* sup  = (float*)d_ws;                                   // 64*256*128
    float* mean = sup + (size_t)BATCH * CROP_N * F_OUT;
    float* rstd = mean + (size_t)CROP_N * F_OUT;

    support_gemm  <<<BATCH * 16, 256, 0, stream>>>(input, weight, sup);
    aggregate_gemm<<<BATCH * 16, 256, 0, stream>>>(adj, sup, out);
    bn_stats      <<<(CROP_N * F_OUT) / 256, 256, 0, stream>>>(out, mean, rstd);
    bn_apply      <<<(BATCH * N_FULL * F_OUT) / 256, 256, 0, stream>>>(out, mean, rstd, gamma, beta);
}